// LikelihoodModel_18253611008687
// MI455X (gfx1250) — compile-verified
//
#include <hip/hip_runtime.h>
#include <cmath>

// Problem constants (from reference)
#define KL1 20
#define KR1 70
#define KL2 120
#define KR2 170
#define KNF 10      // NFACT
#define KT  200     // T
#define KNTR 256    // NTRIALS
#define KNCF 128    // NCONFIGS
#define KTW 50      // TW

// Workspace layout (floats):
//   [0, 2000)          factors = softplus(beta)      (KNF x KT)
//   [2000, 2020)       avg_peak[20]
//   [2020, 2025)       consts: ll1, rl1, ll2, rl2, dt
#define WS_FACT 0
#define WS_PEAK (KNF * KT)
#define WS_CST  (KNF * KT + 2 * KNF)

typedef float vfloat4 __attribute__((ext_vector_type(4)));

#define AS1 __attribute__((address_space(1)))
#define AS3 __attribute__((address_space(3)))

#if defined(__gfx1250__) && __has_builtin(__builtin_amdgcn_global_load_async_to_lds_b32)
#define HAVE_ASYNC_LDS 1
#else
#define HAVE_ASYNC_LDS 0
#endif

__global__ void warp_setup_kernel(const float* __restrict__ beta,
                                  const float* __restrict__ time,
                                  float* __restrict__ ws) {
  float* factors = ws + WS_FACT;
  float* peak    = ws + WS_PEAK;
  float* cst     = ws + WS_CST;

  // softplus(beta), numerically stable: max(x,0) + log1p(exp(-|x|))
  for (int i = threadIdx.x; i < KNF * KT; i += blockDim.x) {
    float x = beta[i];
    factors[i] = fmaxf(x, 0.0f) + log1pf(expf(-fabsf(x)));
  }
  __syncthreads();  // block-level visibility of the global factors writes

  // windowed argmax -> peak times (first-max semantics, like jnp.argmax)
  if (threadIdx.x < 2 * KNF) {
    const int k   = threadIdx.x;
    const int seg = k / KNF;
    const int l   = k % KNF;
    const int L   = seg ? KL2 : KL1;
    float best = -3.402823466e38f;
    int   bi   = 0;
    for (int j = 0; j < (KR1 - KL1); ++j) {
      const float v = factors[l * KT + L + j];
      if (v > best) { best = v; bi = j; }
    }
    peak[k] = time[L + bi];
  }
  if (threadIdx.x == 0) {
    cst[0] = time[KL1];
    cst[1] = time[KR1];
    cst[2] = time[KL2];
    cst[3] = time[KR2];
    cst[4] = time[1] - time[0];
  }
}

// grid: (KNF * (KNTR/4), 2), block: 128 threads.
//   blockIdx.x -> (factor l, 4-trial group nb); blockIdx.y -> time half / segment.
//   tid>>5 selects trial within group; (tid&31)*4 selects float4 config slice.
__global__ void __launch_bounds__(128) warp_main_kernel(
    const float* __restrict__ trial_off,
    const float* __restrict__ config_off,
    const float* __restrict__ ws,
    float* __restrict__ out) {
  __shared__ float F[KT];

  const int tid = threadIdx.x;
  const int l   = blockIdx.x / (KNTR / 4);
  const int nb  = blockIdx.x % (KNTR / 4);
  const int seg = blockIdx.y;            // 0: t in [0,100), 1: t in [100,200)
  const int n   = nb * 4 + (tid >> 5);
  const int c0  = (tid & 31) * 4;

  const float* factors = ws + WS_FACT;
  const float* peakp   = ws + WS_PEAK;
  const float* cst     = ws + WS_CST;

  // Stage this factor's 200-sample row into LDS for the interpolation gathers.
#if HAVE_ASYNC_LDS
  // CDNA5 async data path: GLOBAL_LOAD_ASYNC_TO_LDS_B32 (tracked by ASYNCcnt).
  for (int i = tid; i < KT; i += 128) {
    float* gp = const_cast<float*>(factors + l * KT + i);
    __builtin_amdgcn_global_load_async_to_lds_b32(
        (AS1 int*)gp, (AS3 int*)&F[i], 0, 0);
  }
#if __has_builtin(__builtin_amdgcn_s_wait_asynccnt)
  __builtin_amdgcn_s_wait_asynccnt(0);
#else
  asm volatile("s_wait_asynccnt 0x0" ::: "memory");
#endif
#else
  for (int i = tid; i < KT; i += 128) F[i] = factors[l * KT + i];
#endif

  const int k = seg * KNF + l;

  // gfx1250 global_prefetch_b8 on the strided trial-offset line we need next.
  __builtin_prefetch(&trial_off[(n * KNCF + c0) * (2 * KNF) + k], 0, 0);

  const float dt     = cst[4];
  const float inv_dt = 1.0f / dt;
  const float llv    = cst[seg * 2 + 0];
  const float rlv    = cst[seg * 2 + 1];
  const float pv     = peakp[k];

  // Per-(n,c) warp parameters: hoisted out of the 50-step time loop.
  float lsp[4], lsl[4], rsl[4];
#pragma unroll
  for (int j = 0; j < 4; ++j) {
    const int c = c0 + j;
    float s = pv + trial_off[(n * KNCF + c) * (2 * KNF) + k]
                 + config_off[c * (2 * KNF) + k];
    if (s <= llv) s = llv + dt;
    if (s >= rlv) s = rlv - dt;
    const float L = s - llv;
    lsp[j] = L;
    lsl[j] = (pv - llv) / L;
    rsl[j] = (pv - rlv) / (s - rlv);
  }
  __syncthreads();

  float* obase = out + (((size_t)l * KT) * KNTR + (size_t)n) * KNCF + c0;
  const size_t tstride = (size_t)KNTR * KNCF;

  // Output is written once and never re-read (262 MB > 192 MB L2):
  // stream it with non-temporal hints instead of polluting the cache.
  auto bcast = [&](int ta, int tb) {
    for (int t = ta; t < tb; ++t) {
      const float v = F[t];
      vfloat4 ov = {v, v, v, v};
      __builtin_nontemporal_store(
          ov, reinterpret_cast<vfloat4*>(obase + (size_t)t * tstride));
    }
  };

  auto warp = [&](int tbase) {
    for (int tw = 0; tw < KTW; ++tw) {
      const float lst = (float)tw * dt;
      float o[4];
#pragma unroll
      for (int j = 0; j < 4; ++j) {
        const float wt = (lst < lsp[j])
                             ? (lst * lsl[j] + llv)
                             : ((lst - lsp[j]) * rsl[j] + pv);
        const float wi = wt * inv_dt;
        const float fl = floorf(wi);
        const float cw = wi - fl;
        const int ifl = (int)fl;
        const int ice = (int)ceilf(wi);   // indices stay within [19, 171]
        o[j] = (1.0f - cw) * F[ifl] + cw * F[ice];
      }
      vfloat4 ov = {o[0], o[1], o[2], o[3]};
      __builtin_nontemporal_store(
          ov, reinterpret_cast<vfloat4*>(obase + (size_t)(tbase + tw) * tstride));
    }
  };

  if (seg == 0) {
    bcast(0, KL1);        // t in [0,20): broadcast factors[l,t]
    warp(KL1);            // t in [20,70): warped segment 0
    bcast(KR1, 100);      // t in [70,100): broadcast
  } else {
    bcast(100, KL2);      // t in [100,120): broadcast
    warp(KL2);            // t in [120,170): warped segment 1
    bcast(KR2, KT);       // t in [170,200): broadcast
  }
}

extern "C" void kernel_launch(void* const* d_in, const int* in_sizes, int n_in,
                              void* d_out, int out_size, void* d_ws, size_t ws_size,
                              hipStream_t stream) {
  (void)in_sizes; (void)n_in; (void)out_size; (void)ws_size;
  const float* beta       = (const float*)d_in[0];
  const float* trial_off  = (const float*)d_in[1];
  const float* config_off = (const float*)d_in[2];
  const float* time       = (const float*)d_in[3];
  float* ws  = (float*)d_ws;
  float* out = (float*)d_out;

  warp_setup_kernel<<<1, 256, 0, stream>>>(beta, time, ws);
  warp_main_kernel<<<dim3(KNF * (KNTR / 4), 2), 128, 0, stream>>>(
      trial_off, config_off, ws, out);
}